// SymmetricCPRing_72816875537122
// MI455X (gfx1250) — compile-verified
//
#include <hip/hip_runtime.h>

// SymmetricCPRing on MI455X (gfx1250).
// Factorized: precompute per-element bases (L2-resident), bin atoms by
// element, then per (16-atom tile, channel) fuse nu=3 + nu=2 into ONE
// 16x16x272 GEMM via chained V_WMMA_F32_16X16X4_F32:
//   c1[m,w] = sum_k G[m,k] * Bm[k,w],  G[m,(x,i)] = X[m,x]*X[m,i]
//   Bm rows 0..255 = V3r[(x,i),w],  rows 256..271 = V2r[x,w]
// then c1 += V1[w];  s[m] = sum_w c1[m,w]*X[m,w] (ds_swizzle butterfly);
// out[b,a,c,d] = s[m]*shared[a,c,d].

#define C_DIM 128
#define L_DIM 16
#define E_DIM 10
#define P3_DIM 23
#define P2_DIM 5
#define XT_STRIDE 20  // floats per row: 80B, keeps every b128 dst 16B-aligned

typedef __attribute__((ext_vector_type(2))) float v2f;
typedef __attribute__((ext_vector_type(8))) float v8f;

// ds_swizzle XOR-butterfly add within 16-lane groups (offset[14:10]=xor mask)
#define SWZ_XOR_ADD(p, mask)                                                   \
  p += __int_as_float(                                                         \
      __builtin_amdgcn_ds_swizzle(__float_as_int(p), ((mask) << 10) | 0x1f))

// ---------------------------------------------------------------------------
// Kernel 1: bin atoms by element (y is one-hot [B,E]). Single block.
// ---------------------------------------------------------------------------
__global__ void bin_kernel(const float* __restrict__ y, int B,
                           int* __restrict__ off,        // E+1 prefix counts
                           int* __restrict__ tileStart,  // E+1 prefix tiles
                           int* __restrict__ idxbuf) {   // B atom ids
  __shared__ int cnt[E_DIM];
  __shared__ int cur[E_DIM];
  int t = threadIdx.x;
  if (t < E_DIM) cnt[t] = 0;
  __syncthreads();
  for (int b = t; b < B; b += blockDim.x) {
    const float* yr = y + (size_t)b * E_DIM;
    int e = 0;
    for (int j = 1; j < E_DIM; ++j)
      if (yr[j] > yr[e]) e = j;
    atomicAdd(&cnt[e], 1);
  }
  __syncthreads();
  if (t == 0) {
    int acc = 0, ts = 0;
    for (int e = 0; e < E_DIM; ++e) {
      off[e] = acc;
      tileStart[e] = ts;
      cur[e] = acc;
      ts += (cnt[e] + 15) >> 4;
      acc += cnt[e];
    }
    off[E_DIM] = acc;
    tileStart[E_DIM] = ts;
  }
  __syncthreads();
  for (int b = t; b < B; b += blockDim.x) {
    const float* yr = y + (size_t)b * E_DIM;
    int e = 0;
    for (int j = 1; j < E_DIM; ++j)
      if (yr[j] > yr[e]) e = j;
    int pos = atomicAdd(&cur[e], 1);
    idxbuf[pos] = b;
  }
}

// ---------------------------------------------------------------------------
// Kernel 2: precompute weighted bases per (e,c), stored in WMMA-B-friendly
// [K][N=w] order so B fragments are coalesced b32 loads:
//   V3r[e,c][(x*16+i)*16 + w] = sum_k U3[w,x,i,k] * Wmax[e,k,c]
//   V2r[e,c][x*16 + w]        = sum_k U2[w,x,k]   * W2[e,k,c]
//   V1 [e,c][w]               = U1[w,0] * W1[e,0,c]
// ---------------------------------------------------------------------------
__global__ void __launch_bounds__(256)
precompute_kernel(const float* __restrict__ U3, const float* __restrict__ U2,
                  const float* __restrict__ U1, const float* __restrict__ Wmax,
                  const float* __restrict__ W2, const float* __restrict__ W1,
                  float* __restrict__ V3r, float* __restrict__ V2r,
                  float* __restrict__ V1) {
  int ec = blockIdx.x;
  int e = ec / C_DIM, c = ec % C_DIM;
  __shared__ float wm[P3_DIM];
  __shared__ float w2s[P2_DIM];
  __shared__ float w1s;
  int t = threadIdx.x;
  if (t < P3_DIM) {
    wm[t] = Wmax[((size_t)e * P3_DIM + t) * C_DIM + c];
  } else if (t < P3_DIM + P2_DIM) {
    w2s[t - P3_DIM] = W2[((size_t)e * P2_DIM + (t - P3_DIM)) * C_DIM + c];
  } else if (t == P3_DIM + P2_DIM) {
    w1s = W1[(size_t)e * C_DIM + c];
  }
  __syncthreads();

  float* v3o = V3r + (size_t)ec * 4096;
  #pragma unroll 4
  for (int q0 = 0; q0 < 16; ++q0) {
    int q = t * 16 + q0;  // q = ((w*16+x)*16+i): U3 flat index / P3
    const float* u = U3 + (size_t)q * P3_DIM;
    float s = 0.f;
    #pragma unroll
    for (int k = 0; k < P3_DIM; ++k) s += u[k] * wm[k];
    int w = q >> 8, xx = (q >> 4) & 15, ii = q & 15;
    v3o[(xx * 16 + ii) * 16 + w] = s;  // [K=(x,i)][N=w]
  }
  {
    const float* u = U2 + (size_t)t * P2_DIM;  // t = w*16+x
    float s = 0.f;
    #pragma unroll
    for (int k = 0; k < P2_DIM; ++k) s += u[k] * w2s[k];
    int w = t >> 4, xx = t & 15;
    V2r[(size_t)ec * 256 + xx * 16 + w] = s;  // [K=x][N=w]
  }
  if (t < 16) V1[(size_t)ec * 16 + t] = U1[t] * w1s;
}

// ---------------------------------------------------------------------------
// Kernel 3: main. Block = 128 threads = 4 waves; each wave owns one channel
// c = blockIdx.y*4 + waveId for one 16-atom (same-element) tile.
// ---------------------------------------------------------------------------
__global__ void __launch_bounds__(128)
main_kernel(const float* __restrict__ x, const float* __restrict__ shared_w,
            const float* __restrict__ V3r, const float* __restrict__ V2r,
            const float* __restrict__ V1, const int* __restrict__ off,
            const int* __restrict__ tileStart, const int* __restrict__ idxbuf,
            float* __restrict__ out) {
  __shared__ float xts[4][16 * XT_STRIDE];
  __shared__ int aid[16];

  int tid = blockIdx.x;
  if (tid >= tileStart[E_DIM]) return;  // over-provisioned grid, uniform exit

  int e = 0;
  while (tileStart[e + 1] <= tid) ++e;
  int tileLocal = tid - tileStart[e];
  int rowBase = off[e] + tileLocal * 16;
  int rowEnd = off[e + 1];

  int t = threadIdx.x;
  if (t < 16) {
    int gi = rowBase + t;
    aid[t] = (gi < rowEnd) ? idxbuf[gi] : -1;
  }
  __syncthreads();

  int wave = t >> 5;
  int lane = t & 31;
  int half = lane >> 4;  // 0: lanes 0-15, 1: lanes 16-31
  int l16 = lane & 15;
  int c = blockIdx.y * 4 + wave;
  float* xt = &xts[wave][0];

  // --- async global->LDS copy of the 16x16 x tile (ASYNCcnt path) ---------
  // lane handles row m = lane/2, 8 floats starting at i = (lane&1)*8,
  // as two b128 async copies. Padding rows are clamped then zeroed.
  {
    int m = lane >> 1, i = (lane & 1) * 8;
    int brow = aid[m];
    int bsafe = (brow >= 0) ? brow : idxbuf[rowBase];  // row 0 always valid
    const float* gsrc = x + ((size_t)bsafe * C_DIM + c) * L_DIM + i;
    unsigned ldsa = (unsigned)(size_t)(xt + m * XT_STRIDE + i);
    asm volatile(
        "global_load_async_to_lds_b128 %0, %1, off\n\t"
        "global_load_async_to_lds_b128 %0, %1, off offset:16"
        :
        : "v"(ldsa), "v"(gsrc)
        : "memory");
    asm volatile("s_wait_asynccnt 0x0" ::: "memory");
    if (brow < 0) {
      #pragma unroll
      for (int j = 0; j < 8; ++j) xt[m * XT_STRIDE + i + j] = 0.f;
    }
  }

  // lane-private copy of its row (constant-indexed => stays in VGPRs)
  float xr[16];
  #pragma unroll
  for (int i = 0; i < 16; ++i) xr[i] = xt[l16 * XT_STRIDE + i];

  const float* v3b = V3r + ((size_t)e * C_DIM + c) * 4096;
  const float* v2b = V2r + ((size_t)e * C_DIM + c) * 256;
  const float* v1b = V1 + ((size_t)e * C_DIM + c) * 16;
  __builtin_prefetch(v3b, 0, 0);  // global_prefetch_b8

  // --- fused 16x16x272 GEMM: c1 = G @ [V3r; V2r] --------------------------
  // A layout (f32 16x4): VGPR0 lanes0-15 K=4s, lanes16-31 K=4s+2; VGPR1 +1.
  // B layout: VGPR0 lanes = N=w for row K=4s(+2*half); VGPR1 row K+1.
  v8f acc = {};
  #pragma unroll
  for (int s = 0; s < 64; ++s) {  // K rows 0..255: k = x*16+i
    const int xc = (4 * s) >> 4;  // x index (same for k..k+3)
    const int ic = (4 * s) & 15;  // i base
    float ax = xr[xc];
    v2f afrag;
    afrag.x = ax * (half ? xr[ic + 2] : xr[ic + 0]);
    afrag.y = ax * (half ? xr[ic + 3] : xr[ic + 1]);
    const float* bp = v3b + (4 * s + half * 2) * 16 + l16;
    v2f bfrag;
    bfrag.x = bp[0];   // row k   (coalesced b32 across lanes)
    bfrag.y = bp[16];  // row k+1
    acc = __builtin_amdgcn_wmma_f32_16x16x4_f32(false, afrag, false, bfrag,
                                                (short)0, acc, false, false);
  }
  #pragma unroll
  for (int s = 0; s < 4; ++s) {  // K rows 256..271: nu=2 term, A = X[m,x]
    const int x0 = 4 * s;
    v2f afrag;
    afrag.x = (half ? xr[x0 + 2] : xr[x0 + 0]);
    afrag.y = (half ? xr[x0 + 3] : xr[x0 + 1]);
    const float* bp = v2b + (x0 + half * 2) * 16 + l16;
    v2f bfrag;
    bfrag.x = bp[0];
    bfrag.y = bp[16];
    acc = __builtin_amdgcn_wmma_f32_16x16x4_f32(false, afrag, false, bfrag,
                                                (short)0, acc, false, false);
  }

  // --- nu=1 + final dot over w + output mix -------------------------------
  float v1v = v1b[l16];  // V1[w], w = l16 (same for both lane halves)
  float sw00 = shared_w[(size_t)c * 2 + 0];
  float sw01 = shared_w[(size_t)c * 2 + 1];
  float sw10 = shared_w[((size_t)C_DIM + c) * 2 + 0];
  float sw11 = shared_w[((size_t)C_DIM + c) * 2 + 1];

  #pragma unroll
  for (int r = 0; r < 8; ++r) {
    int m = r + half * 8;  // C/D layout: VGPR r -> rows {r, r+8}
    float p = (acc[r] + v1v) * xt[m * XT_STRIDE + l16];  // c1[m,w]*X[m,w]
    SWZ_XOR_ADD(p, 1);
    SWZ_XOR_ADD(p, 2);
    SWZ_XOR_ADD(p, 4);
    SWZ_XOR_ADD(p, 8);  // all lanes in each 16-group now hold s[m]
    if (l16 == 0) {
      int b = aid[m];
      if (b >= 0) {
        float* o = out + (((size_t)b * 2) * C_DIM + c) * 2;
        o[0] = p * sw00;
        o[1] = p * sw01;
        o[(size_t)C_DIM * 2 + 0] = p * sw10;
        o[(size_t)C_DIM * 2 + 1] = p * sw11;
      }
    }
  }
}

// ---------------------------------------------------------------------------
extern "C" void kernel_launch(void* const* d_in, const int* in_sizes, int n_in,
                              void* d_out, int out_size, void* d_ws,
                              size_t ws_size, hipStream_t stream) {
  const float* x = (const float*)d_in[0];
  const float* y = (const float*)d_in[1];
  const float* U3 = (const float*)d_in[2];
  const float* U2 = (const float*)d_in[3];
  const float* U1 = (const float*)d_in[4];
  const float* Wmax = (const float*)d_in[5];
  const float* W2 = (const float*)d_in[6];
  const float* W1 = (const float*)d_in[7];
  const float* sharedw = (const float*)d_in[8];
  float* out = (float*)d_out;

  int B = in_sizes[0] / (C_DIM * L_DIM);  // 4096

  // Workspace: V3r (21.0 MB) | V2r (1.31 MB) | V1 (82 KB) | ints
  size_t nV3 = (size_t)E_DIM * C_DIM * 4096;
  size_t nV2 = (size_t)E_DIM * C_DIM * 256;
  size_t nV1 = (size_t)E_DIM * C_DIM * 16;
  float* V3r = (float*)d_ws;
  float* V2r = V3r + nV3;
  float* V1 = V2r + nV2;
  int* off = (int*)(V1 + nV1);
  int* tileStart = off + (E_DIM + 1);
  int* idxbuf = tileStart + (E_DIM + 1);

  hipLaunchKernelGGL(bin_kernel, dim3(1), dim3(256), 0, stream, y, B, off,
                     tileStart, idxbuf);
  hipLaunchKernelGGL(precompute_kernel, dim3(E_DIM * C_DIM), dim3(256), 0,
                     stream, U3, U2, U1, Wmax, W2, W1, V3r, V2r, V1);
  int maxTiles = B / 16 + E_DIM;  // upper bound on sum of per-bucket tiles
  hipLaunchKernelGGL(main_kernel, dim3(maxTiles, C_DIM / 4), dim3(128), 0,
                     stream, x, sharedw, V3r, V2r, V1, off, tileStart, idxbuf,
                     out);
}